// channel_grouping_loss_54735063220842
// MI455X (gfx1250) — compile-verified
//
#include <hip/hip_runtime.h>
#include <stdint.h>

// Problem shape (fixed by the reference): feature [B=32, C=64, H=128, W=128] fp32.
#define BB 32
#define CC 64
#define HH 128
#define WW 128
#define HW (HH * WW)        // 16384 elements per (b,c) map = 64 KiB
#define NMAPS (BB * CC)     // 2048
#define NPIX (BB * HW)      // 524288

typedef __attribute__((ext_vector_type(2))) float v2f;
typedef __attribute__((ext_vector_type(8))) float v8f;
typedef __attribute__((ext_vector_type(4))) unsigned int u32x4;
typedef __attribute__((ext_vector_type(8))) int i32x8;
typedef __attribute__((ext_vector_type(4))) int i32x4;

__device__ inline float wave_sum_f(float x) {
#pragma unroll
  for (int o = 16; o > 0; o >>= 1) x += __shfl_xor(x, o, 32);
  return x;
}
__device__ inline double wave_sum_d(double x) {
#pragma unroll
  for (int o = 16; o > 0; o >>= 1) x += __shfl_xor(x, o, 32);
  return x;
}

// ---------------------------------------------------------------------------
// Pass 1: one workgroup per (b,c) 128x128 map.
//   - Stage the 64KB tile into LDS with the Tensor Data Mover (TDM), 6-arg
//     clang-23 builtin form; fall back to a plain staged copy if unavailable.
//   - Row sums of g=f^2 via v_wmma_f32_16x16x4_f32:  D = A(16x4) * ones(4x16)
//   - Col sums of g     via v_wmma_f32_16x16x4_f32:  D = ones(16x4) * B(4x16)
//   - First-occurrence argmax + sum(f) on the VALU while building A fragments.
//   Writes per-block partials: dis contribution and sum(f).
// ---------------------------------------------------------------------------
__global__ __launch_bounds__(256) void cg_pass1(const float* __restrict__ feat,
                                                double* __restrict__ dis_part,
                                                double* __restrict__ sumf_part) {
  __shared__ float tile[HW];
  __shared__ double wsum[8][6];  // s0, si, sii, sj, sjj, sumf per wave
  __shared__ float wbv[8];
  __shared__ int wbi[8];

  const int map = blockIdx.x;  // b*C + c
  const int tid = threadIdx.x;
  const int lane = tid & 31;
  const int wv = tid >> 5;
  const float* __restrict__ src = feat + (size_t)map * HW;

  // ---- stage tile (global -> LDS) ----
#if __has_builtin(__builtin_amdgcn_tensor_load_to_lds)
  if (wv == 0) {
    const unsigned lds_off = (unsigned)(uintptr_t)tile;  // low 32 bits of generic = LDS byte offset
    const unsigned long long ga = (unsigned long long)(uintptr_t)src;
    u32x4 g0;
    g0[0] = 1u;                                              // count=1, user descriptor
    g0[1] = lds_off;                                         // lds_addr
    g0[2] = (unsigned)(ga & 0xffffffffu);                    // global_addr[31:0]
    g0[3] = (unsigned)((ga >> 32) & 0x1ffffffu) | (2u << 30);// global_addr[56:32] | type=2
    i32x8 g1;
    g1[0] = 0x20000;                 // data_size=2 (4 bytes), no multicast/pad/iterate
    g1[1] = (int)(128u << 16);       // tensor_dim0[15:0] in bits 63:48
    g1[2] = (int)(128u << 16);       // tensor_dim1[15:0] in bits 95:80 (dim0 hi = 0)
    g1[3] = (int)(128u << 16);       // tile_dim0=128 in bits 127:112 (dim1 hi = 0)
    g1[4] = 128;                     // tile_dim1=128, tile_dim2=0
    g1[5] = 128;                     // tensor_dim0_stride[31:0] = 128 elements
    g1[6] = (int)(16384u << 16);     // tensor_dim1_stride[15:0] (hi bits of stride0 = 0)
    g1[7] = 0;                       // tensor_dim1_stride[47:16] = 0
    i32x4 gz4 = {0, 0, 0, 0};        // groups 2/3 unused (2D tile)
    i32x8 gz8 = {0, 0, 0, 0, 0, 0, 0, 0};
    __builtin_amdgcn_tensor_load_to_lds(g0, g1, gz4, gz4, gz8, 0);
#if __has_builtin(__builtin_amdgcn_s_wait_tensorcnt)
    __builtin_amdgcn_s_wait_tensorcnt(0);
#else
    asm volatile("s_wait_tensorcnt 0x0" ::: "memory");
#endif
  }
  __syncthreads();
#else
  {
    const float4* __restrict__ s4 = (const float4*)src;
    float4* t4 = (float4*)tile;
    for (int i = tid; i < HW / 4; i += 256) t4[i] = s4[i];
  }
  __syncthreads();
#endif

  const int m = lane & 15;   // row (A) / column (B) within the 16-wide fragment
  const int kh = lane >> 4;  // which K-half this lane holds

  v2f ones;
  ones.x = 1.0f;
  ones.y = 1.0f;

  // ---- row pass: D += A(g) x ones ; also argmax + sum(f) ----
  v8f accR = {0.f, 0.f, 0.f, 0.f, 0.f, 0.f, 0.f, 0.f};
  float sumf = 0.0f;
  float bestv = -3.402823466e38f;
  int besti = 0;
  const int row = wv * 16 + m;  // 8 waves x 16 rows = 128 rows
  const float* trow = tile + row * WW;
#pragma unroll 4
  for (int it = 0; it < 32; ++it) {
    const int j0 = it * 4 + kh * 2;  // this lane's two contiguous columns
    const float f0 = trow[j0];
    const float f1 = trow[j0 + 1];
    sumf += f0 + f1;
    const int id0 = row * WW + j0;
    if (f0 > bestv) { bestv = f0; besti = id0; }      // per-lane idx increases -> strict '>'
    if (f1 > bestv) { bestv = f1; besti = id0 + 1; }  // keeps first occurrence
    v2f a;
    a.x = f0 * f0;
    a.y = f1 * f1;
    accR = __builtin_amdgcn_wmma_f32_16x16x4_f32(false, a, false, ones, (short)0,
                                                 accR, false, false);
  }
  // D layout: VGPR v holds rowsum(16w+v) in lanes 0-15, rowsum(16w+8+v) in 16-31,
  // replicated across the 16 columns -> keep only column n==0 of each half.
  float s0 = 0.f, si = 0.f, sii = 0.f;
  {
    const int mb = wv * 16 + (kh ? 8 : 0);
#pragma unroll
    for (int v = 0; v < 8; ++v) {
      const float r = accR[v];
      const float fi = (float)(mb + v);
      s0 += r;
      si += r * fi;
      sii += r * fi * fi;
    }
    if (m != 0) { s0 = 0.f; si = 0.f; sii = 0.f; }
  }

  // ---- col pass: D += ones x B(g) ----
  v8f accC = {0.f, 0.f, 0.f, 0.f, 0.f, 0.f, 0.f, 0.f};
  const int cj = wv * 16 + m;  // 8 waves x 16 cols = 128 cols
#pragma unroll 4
  for (int it = 0; it < 32; ++it) {
    const int r0 = it * 4 + kh * 2;  // this lane's two rows of the 4-row chunk
    const float g0 = tile[r0 * WW + cj];
    const float g1 = tile[(r0 + 1) * WW + cj];
    v2f bfr;
    bfr.x = g0 * g0;
    bfr.y = g1 * g1;
    accC = __builtin_amdgcn_wmma_f32_16x16x4_f32(false, ones, false, bfr, (short)0,
                                                 accC, false, false);
  }
  // D[m,n] = colsum(16w + n), identical in every VGPR; lanes 16-31 duplicate lanes 0-15.
  float sj = 0.f, sjj = 0.f;
  if (!kh) {
    const float cs = accC[0];
    const float fj = (float)cj;
    sj = cs * fj;
    sjj = sj * fj;
  }

  // ---- wave reduction ----
  s0 = wave_sum_f(s0);
  si = wave_sum_f(si);
  sii = wave_sum_f(sii);
  sj = wave_sum_f(sj);
  sjj = wave_sum_f(sjj);
  sumf = wave_sum_f(sumf);
#pragma unroll
  for (int o = 16; o > 0; o >>= 1) {
    const float ov = __shfl_xor(bestv, o, 32);
    const int oi = __shfl_xor(besti, o, 32);
    if (ov > bestv || (ov == bestv && oi < besti)) { bestv = ov; besti = oi; }
  }
  if (lane == 0) {
    wsum[wv][0] = (double)s0;
    wsum[wv][1] = (double)si;
    wsum[wv][2] = (double)sii;
    wsum[wv][3] = (double)sj;
    wsum[wv][4] = (double)sjj;
    wsum[wv][5] = (double)sumf;
    wbv[wv] = bestv;
    wbi[wv] = besti;
  }
  __syncthreads();
  if (tid == 0) {
    double S0 = 0, Si = 0, Sii = 0, Sj = 0, Sjj = 0, SF = 0;
    float bv = -3.402823466e38f;
    int bi = 0x7fffffff;
#pragma unroll
    for (int w = 0; w < 8; ++w) {
      S0 += wsum[w][0];
      Si += wsum[w][1];
      Sii += wsum[w][2];
      Sj += wsum[w][3];
      Sjj += wsum[w][4];
      SF += wsum[w][5];
      if (wbv[w] > bv || (wbv[w] == bv && wbi[w] < bi)) { bv = wbv[w]; bi = wbi[w]; }
    }
    const double mx = (double)(bi / WW);
    const double my = (double)(bi % WW);
    const double dis = mx * mx * S0 - 2.0 * mx * Si + Sii +
                       my * my * S0 - 2.0 * my * Sj + Sjj;
    dis_part[map] = dis;
    sumf_part[map] = SF;
  }
}

// ---------------------------------------------------------------------------
// Pass 2: one thread per pixel (b,h,w). Scan C=64 channels (coalesced across w):
// track m1 (max), m2 (2nd max), S2 = sum f^2. Per-pixel closed forms remove the
// need for argmax channel or a re-read:
//   A += m1^2 (S2 - m1^2) + m1^2 m2^2
//   B += m1   (S2 - m1^2) + m2 m1^2
//   C += S2
// div_loss = A - 2 mgr B + mgr^2 C  folded in at finalize.
// ---------------------------------------------------------------------------
__global__ __launch_bounds__(256) void cg_pass2(const float* __restrict__ feat,
                                                double* __restrict__ a_part,
                                                double* __restrict__ b_part,
                                                double* __restrict__ c_part) {
  __shared__ double sA[8], sB[8], sC[8];
  const int tid = threadIdx.x;
  const int g = blockIdx.x * 256 + tid;  // 0 .. NPIX-1
  const int b = g >> 14;                 // / (H*W)
  const int hw = g & (HW - 1);
  const float* __restrict__ p = feat + (size_t)b * CC * HW + hw;

  float m1 = -3.402823466e38f, m2 = -3.402823466e38f, s2 = 0.f;
#pragma unroll 8
  for (int c = 0; c < CC; ++c) {
    const float v = p[(size_t)c * HW];
    s2 = fmaf(v, v, s2);
    if (v > m1) {
      m2 = m1;
      m1 = v;
    } else if (v > m2) {
      m2 = v;
    }
  }
  const float m1sq = m1 * m1;
  const float rest = s2 - m1sq;
  double Ap = (double)m1sq * (double)rest + (double)(m2 * m2) * (double)m1sq;
  double Bp = (double)m1 * (double)rest + (double)m2 * (double)m1sq;
  double Cp = (double)s2;

  Ap = wave_sum_d(Ap);
  Bp = wave_sum_d(Bp);
  Cp = wave_sum_d(Cp);
  const int lane = tid & 31;
  const int wv = tid >> 5;
  if (lane == 0) {
    sA[wv] = Ap;
    sB[wv] = Bp;
    sC[wv] = Cp;
  }
  __syncthreads();
  if (tid == 0) {
    double A = 0, B = 0, C = 0;
#pragma unroll
    for (int w = 0; w < 8; ++w) {
      A += sA[w];
      B += sB[w];
      C += sC[w];
    }
    a_part[blockIdx.x] = A;
    b_part[blockIdx.x] = B;
    c_part[blockIdx.x] = C;
  }
}

// ---------------------------------------------------------------------------
// Pass 3: single block, deterministic fixed-order combine of 2048 partials each.
// ---------------------------------------------------------------------------
__global__ __launch_bounds__(256) void cg_finalize(const double* __restrict__ ws,
                                                   float* __restrict__ out) {
  __shared__ double sm[8][5];
  const int tid = threadIdx.x;
  double a0 = 0, a1 = 0, a2 = 0, a3 = 0, a4 = 0;
  for (int i = tid; i < NMAPS; i += 256) {
    a0 += ws[i];              // dis partials
    a1 += ws[NMAPS + i];      // sum(f) partials
    a2 += ws[2 * NMAPS + i];  // A partials
    a3 += ws[3 * NMAPS + i];  // B partials
    a4 += ws[4 * NMAPS + i];  // C partials
  }
  a0 = wave_sum_d(a0);
  a1 = wave_sum_d(a1);
  a2 = wave_sum_d(a2);
  a3 = wave_sum_d(a3);
  a4 = wave_sum_d(a4);
  const int lane = tid & 31;
  const int wv = tid >> 5;
  if (lane == 0) {
    sm[wv][0] = a0;
    sm[wv][1] = a1;
    sm[wv][2] = a2;
    sm[wv][3] = a3;
    sm[wv][4] = a4;
  }
  __syncthreads();
  if (tid == 0) {
    double D = 0, SF = 0, A = 0, B = 0, C = 0;
#pragma unroll
    for (int w = 0; w < 8; ++w) {
      D += sm[w][0];
      SF += sm[w][1];
      A += sm[w][2];
      B += sm[w][3];
      C += sm[w][4];
    }
    const double n = (double)BB * CC * HH * WW;  // 33554432
    const double mgr = SF / n;
    out[0] = (float)(D / n);
    out[1] = (float)((A - 2.0 * mgr * B + mgr * mgr * C) / n);
  }
}

extern "C" void kernel_launch(void* const* d_in, const int* in_sizes, int n_in,
                              void* d_out, int out_size, void* d_ws, size_t ws_size,
                              hipStream_t stream) {
  (void)in_sizes;
  (void)n_in;
  (void)out_size;
  (void)ws_size;  // needs 5 * 2048 doubles = 81920 bytes of scratch
  const float* feat = (const float*)d_in[0];
  float* out = (float*)d_out;
  double* ws = (double*)d_ws;
  double* dis_part = ws;                // [2048]
  double* sumf_part = ws + NMAPS;       // [2048]
  double* a_part = ws + 2 * NMAPS;      // [2048]
  double* b_part = ws + 3 * NMAPS;      // [2048]
  double* c_part = ws + 4 * NMAPS;      // [2048]

  cg_pass1<<<NMAPS, 256, 0, stream>>>(feat, dis_part, sumf_part);
  cg_pass2<<<NPIX / 256, 256, 0, stream>>>(feat, a_part, b_part, c_part);
  cg_finalize<<<1, 256, 0, stream>>>(ws, out);
}